// SegmentPatternStage_68676527063196
// MI455X (gfx1250) — compile-verified
//
#include <hip/hip_runtime.h>

// Problem constants (from reference)
#define B_      64
#define CIN_    128
#define LIN_    3000
#define COUT_   64
#define K_      9
#define STRIDE_ 2
#define PAD_    4
#define LOUT_   1500      // (3000 + 8 - 9)/2 + 1
#define T_      25
#define BETA_   0.9f
#define THR_    1.0f

#define LTILE   32                              // output positions per block
#define NLBLK   ((LOUT_ + LTILE - 1) / LTILE)   // 47 blocks along L
#define ZSTR    33                              // padded LDS row stride (floats)

typedef float v2f __attribute__((ext_vector_type(2)));
typedef float v8f __attribute__((ext_vector_type(8)));

// Fused: depthwise conv -> (LDS) -> pointwise GEMM via v_wmma_f32_16x16x4_f32 -> LIF -> rate
__global__ __launch_bounds__(256)
void dsconv_lif_kernel(const float* __restrict__ x,
                       const float* __restrict__ dw,
                       const float* __restrict__ pw,
                       float* __restrict__ out,
                       float* __restrict__ partial)
{
    __shared__ float z[CIN_ * ZSTR];   // depthwise result tile: 128 channels x 32 positions
    __shared__ float red[256];

    const int tid  = threadIdx.x;
    const int lane = tid & 31;
    const int wave = tid >> 5;         // 8 waves
    const int lblk = blockIdx.x;       // 0..46
    const int b    = blockIdx.y;       // 0..63
    const int l0   = lblk * LTILE;

    // ---------------- Stage 1: depthwise conv ----------------
    // wave w handles channels [16w, 16w+16); lane <-> output position l0+lane
    {
        const int ibase = (l0 + lane) * STRIDE_ - PAD_;
        for (int i = 0; i < 16; ++i) {
            const int ci = wave * 16 + i;
            const float* __restrict__ xrow = x + ((size_t)b * CIN_ + ci) * LIN_;
            const float* __restrict__ wrow = dw + ci * K_;
            float acc = 0.0f;
            #pragma unroll
            for (int j = 0; j < K_; ++j) {
                const int idx = ibase + j;
                const float xv = (idx >= 0 && idx < LIN_) ? xrow[idx] : 0.0f;
                acc = fmaf(xv, wrow[j], acc);
            }
            z[ci * ZSTR + lane] = acc;
        }
    }
    __syncthreads();

    // ---------------- Stage 2: pointwise GEMM (WMMA fp32) ----------------
    // wave -> (co tile, l tile): 4 co tiles x 2 l tiles of 16x16
    const int co_t   = wave & 3;
    const int l_t    = wave >> 2;
    const int grp    = lane >> 4;      // 0: K{0,1}/M rows 0-7 ; 1: K{2,3}/M rows 8-15
    const int mn     = lane & 15;      // A row (m) / B-C column (n)
    const int cobase = co_t * 16;

    v8f c = {0.f,0.f,0.f,0.f,0.f,0.f,0.f,0.f};
    #pragma unroll
    for (int kk = 0; kk < CIN_ / 4; ++kk) {
        const int k2 = kk * 4 + grp * 2;
        // A: pw[co, ci] -- 16x4 f32 A layout (lane = grp*16 + m, vgprs hold k2, k2+1)
        const float* __restrict__ ap = pw + (size_t)(cobase + mn) * CIN_ + k2;
        v2f a;
        a.x = ap[0];
        a.y = ap[1];
        // B: z[ci, l] -- 4x16 f32 B layout (lane = grp*16 + n)
        v2f bb;
        bb.x = z[(k2    ) * ZSTR + l_t * 16 + mn];
        bb.y = z[(k2 + 1) * ZSTR + l_t * 16 + mn];
        c = __builtin_amdgcn_wmma_f32_16x16x4_f32(
                /*neg_a=*/false, a, /*neg_b=*/false, bb,
                /*c_mod=*/(short)0, c, /*reuse_a=*/false, /*reuse_b=*/false);
    }

    // ---------------- Stage 3: LIF on accumulators ----------------
    float lsum = 0.0f;
    const int l = l0 + l_t * 16 + mn;
    #pragma unroll
    for (int r = 0; r < 8; ++r) {
        const float cur = c[r];
        float mem = 0.0f, acc = 0.0f;
        #pragma unroll
        for (int t = 0; t < T_; ++t) {
            mem = fmaf(mem, BETA_, cur);          // mem = beta*mem + cur
            const float s = (mem > THR_) ? 1.0f : 0.0f;
            acc += s;
            mem -= s * THR_;                      // soft reset
        }
        const float rate = acc * (1.0f / (float)T_);
        if (l < LOUT_) {
            const int co = cobase + grp * 8 + r;  // C layout: vgpr r, rows r / r+8
            out[((size_t)b * COUT_ + co) * LOUT_ + l] = rate;
            lsum += rate;
        }
    }

    // ---------------- per-block partial sum (deterministic) ----------------
    red[tid] = lsum;
    __syncthreads();
    #pragma unroll
    for (int s = 128; s > 0; s >>= 1) {
        if (tid < s) red[tid] += red[tid + s];
        __syncthreads();
    }
    if (tid == 0) partial[blockIdx.y * gridDim.x + blockIdx.x] = red[0];
}

// Fold per-block partials into the scalar regularizer, fixed order -> deterministic.
__global__ __launch_bounds__(256)
void reg_reduce_kernel(const float* __restrict__ partial, int n,
                       float* __restrict__ out_scalar)
{
    __shared__ float red[256];
    float s = 0.0f;
    for (int i = threadIdx.x; i < n; i += 256) s += partial[i];
    red[threadIdx.x] = s;
    __syncthreads();
    #pragma unroll
    for (int k = 128; k > 0; k >>= 1) {
        if (threadIdx.x < k) red[threadIdx.x] += red[threadIdx.x + k];
        __syncthreads();
    }
    if (threadIdx.x == 0)
        *out_scalar = red[0] / (float)((size_t)B_ * COUT_ * LOUT_);
}

extern "C" void kernel_launch(void* const* d_in, const int* in_sizes, int n_in,
                              void* d_out, int out_size, void* d_ws, size_t ws_size,
                              hipStream_t stream)
{
    const float* x  = (const float*)d_in[0];   // (64,128,3000)
    const float* dw = (const float*)d_in[1];   // (128,1,9)
    const float* pw = (const float*)d_in[2];   // (64,128,1)
    float* out      = (float*)d_out;           // 64*64*1500 rates + 1 scalar
    float* partial  = (float*)d_ws;            // 47*64 partial sums (12 KB)

    dim3 grid(NLBLK, B_);
    dsconv_lif_kernel<<<grid, 256, 0, stream>>>(x, dw, pw, out, partial);
    reg_reduce_kernel<<<1, 256, 0, stream>>>(partial, NLBLK * B_,
                                             out + (size_t)B_ * COUT_ * LOUT_);
}